// LoRALinear_10703058502148
// MI455X (gfx1250) — compile-verified
//
#include <hip/hip_runtime.h>
#include <hip/hip_bf16.h>

typedef __attribute__((ext_vector_type(2))) float v2f;
typedef __attribute__((ext_vector_type(8))) float v8f;

// ---------------------------------------------------------------------------
// Stage 1: T[16384,16] = x[16384,4096] @ lora_A[16,4096]^T
// One block (8 waves) per 16-row M tile. K=4096 split across 8 waves
// (512 each), partial 16x16 accumulators reduced via LDS (deterministic).
// ---------------------------------------------------------------------------
__global__ __launch_bounds__(256) void lora_stage1(
    const float* __restrict__ x,   // [16384, 4096]
    const float* __restrict__ A,   // [16, 4096]
    float* __restrict__ T)         // [16384, 16]
{
    __shared__ float red[8 * 256];

    const int lane  = threadIdx.x & 31;
    const int wave  = threadIdx.x >> 5;
    const int tile  = blockIdx.x;            // 0..1023  (M tile)
    const int mrow  = (tile << 4) + (lane & 15);
    const int koff  = (lane >> 4) << 1;      // lanes 0-15 -> K+0,1 ; 16-31 -> K+2,3
    const int kbase = wave << 9;             // 512 K per wave

    const float* xp = x + (size_t)mrow * 4096 + kbase + koff;          // WMMA A frag
    const float* ap = A + (size_t)(lane & 15) * 4096 + kbase + koff;   // WMMA B frag (B[k][n]=A[n][k])

    v8f c0 = {}, c1 = {};
#pragma unroll 4
    for (int kk = 0; kk < 512; kk += 8) {
        v2f a0 = __builtin_nontemporal_load((const v2f*)(xp + kk));      // x streamed once
        v2f b0 = *(const v2f*)(ap + kk);                                 // A stays in cache
        v2f a1 = __builtin_nontemporal_load((const v2f*)(xp + kk + 4));
        v2f b1 = *(const v2f*)(ap + kk + 4);
        c0 = __builtin_amdgcn_wmma_f32_16x16x4_f32(false, a0, false, b0,
                                                   (short)0, c0, false, false);
        c1 = __builtin_amdgcn_wmma_f32_16x16x4_f32(false, a1, false, b1,
                                                   (short)0, c1, false, false);
    }
    v8f c = c0 + c1;

    // Stash this wave's partial tile in LDS.
#pragma unroll
    for (int r = 0; r < 8; ++r)
        red[wave * 256 + r * 32 + lane] = c[r];
    __syncthreads();

    // Deterministic cross-wave reduction: thread t sums slot t of all 8 waves.
    float s = 0.f;
#pragma unroll
    for (int w = 0; w < 8; ++w)
        s += red[w * 256 + threadIdx.x];

    // Map flat slot -> (reg r, lane l) -> C/D element (M = r + 8*(l>=16), N = l&15).
    const int r  = threadIdx.x >> 5;
    const int l  = threadIdx.x & 31;
    const int mm = r + ((l >> 4) << 3);
    const int nn = l & 15;
    T[((size_t)(tile << 4) + mm) * 16 + nn] = s;
}

// ---------------------------------------------------------------------------
// Stage 2: out[16384,4096] = T[16384,16] @ lora_B[4096,16]^T
// One wave per 16x16 output tile; K=16 -> 4 chained WMMAs.
// T (1MB) and lora_B (256KB) remain L2-resident; out is written NT.
// ---------------------------------------------------------------------------
__global__ __launch_bounds__(256) void lora_stage2(
    const float* __restrict__ T,   // [16384, 16]
    const float* __restrict__ Bm,  // [4096, 16]
    float* __restrict__ out)       // [16384, 4096]
{
    const int lane = threadIdx.x & 31;
    const int wave = threadIdx.x >> 5;
    const int mt   = blockIdx.x;                 // 0..1023
    const int nt   = (blockIdx.y << 3) + wave;   // 0..255
    const int mrow = (mt << 4) + (lane & 15);
    const int nrow = (nt << 4) + (lane & 15);
    const int koff = (lane >> 4) << 1;

    const float* tp = T  + (size_t)mrow * 16 + koff;   // A frag: T tile rows
    const float* bp = Bm + (size_t)nrow * 16 + koff;   // B frag: B^T[k][n] = lora_B[n][k]

    v8f c = {};
#pragma unroll
    for (int s = 0; s < 4; ++s) {
        v2f a = *(const v2f*)(tp + 4 * s);
        v2f b = *(const v2f*)(bp + 4 * s);
        c = __builtin_amdgcn_wmma_f32_16x16x4_f32(false, a, false, b,
                                                  (short)0, c, false, false);
    }

    // C/D layout: reg r, lanes 0-15 -> row r, lanes 16-31 -> row r+8.
    float* op = out + ((size_t)(mt << 4) + ((lane >> 4) << 3)) * 4096
                    + (size_t)(nt << 4) + (lane & 15);
#pragma unroll
    for (int r = 0; r < 8; ++r)
        __builtin_nontemporal_store(c[r], op + (size_t)r * 4096);
}

// ---------------------------------------------------------------------------
extern "C" void kernel_launch(void* const* d_in, const int* in_sizes, int n_in,
                              void* d_out, int out_size, void* d_ws, size_t ws_size,
                              hipStream_t stream) {
    const float* x  = (const float*)d_in[0];   // [4,4096,4096] -> [16384,4096]
    const float* A  = (const float*)d_in[1];   // [16,4096]
    const float* Bm = (const float*)d_in[2];   // [4096,16]
    float* out      = (float*)d_out;           // [16384,4096]
    float* T        = (float*)d_ws;            // [16384,16] = 1 MB scratch

    // Stage 1: 1024 M-tiles, 8 waves each (K split), deterministic LDS reduce.
    lora_stage1<<<dim3(1024), dim3(256), 0, stream>>>(x, A, T);

    // Stage 2: 1024 x 256 output tiles, 8 tiles per block.
    lora_stage2<<<dim3(1024, 32), dim3(256), 0, stream>>>(T, Bm, out);
}